// GNNStack_53137335386502
// MI455X (gfx1250) — compile-verified
//
#include <hip/hip_runtime.h>

// ---------------------------------------------------------------------------
// GNNStack on MI455X (gfx1250, wave32). All matmuls but the decoder's 128->3 via V_WMMA_F32_16X16X4_F32
// (exact fp32). One wave owns one 16-row tile.
//
// Workspace layout (floats): h0[N*64] | h1[N*64] | agg[N*64]  => ~38.4 MB.
// ---------------------------------------------------------------------------

typedef float v2f __attribute__((ext_vector_type(2)));
typedef float v8f __attribute__((ext_vector_type(8)));

#define NN      50000
#define EE      800000
#define EMB     64
#define HID     128
#define MREAL   25000
#define NTILE_N 3125      // NN / 16 (exact)
#define NTILE_M 1563      // ceil(MREAL / 16)

__device__ __forceinline__ v8f wmma4(v2f a, v2f b, v8f c) {
  // D(16x16,f32) = A(16x4,f32) * B(4x16,f32) + C
  return __builtin_amdgcn_wmma_f32_16x16x4_f32(
      /*neg_a=*/false, a, /*neg_b=*/false, b,
      /*c_mod=*/(short)0, c, /*reuse_a=*/false, /*reuse_b=*/false);
}

__device__ __forceinline__ v8f bcast8(float v) {
  v8f r;
#pragma unroll
  for (int i = 0; i < 8; ++i) r[i] = v;
  return r;
}

// B-fragment: W is row-major [K x ldw]; tile cols n0..n0+15, k-slice k0..k0+3.
// Layout: v0: lanes0-15 K=k0 / lanes16-31 K=k0+2 ; v1: K=k0+1 / K=k0+3.
__device__ __forceinline__ v2f bfrag(const float* W, int ldw, int k0, int n0, int lane) {
  int kk  = k0 + 2 * (lane >> 4);
  int col = n0 + (lane & 15);
  v2f b;
  b.x = W[kk * ldw + col];
  b.y = W[(kk + 1) * ldw + col];
  return b;
}

// A-fragment from a row-major [rows x ld] matrix (consecutive K pair per lane).
__device__ __forceinline__ v2f afrag_g(const float* A, int ld, int row0, int k0, int lane) {
  int r  = row0 + (lane & 15);
  int kk = k0 + 2 * (lane >> 4);
  float2 t = *(const float2*)(A + (size_t)r * ld + kk);
  v2f a; a.x = t.x; a.y = t.y; return a;
}

__device__ __forceinline__ float prelu(float v, float s) { return v > 0.0f ? v : s * v; }

// ---------------------------------------------------------------------------
// Encoder: h0[n,64] = MLP(concat(x[n,16], u[batch[n],8]))  (24->128->128->64)
// 128 threads = 4 waves; wave = one 16-node tile; per-wave 16x128 LDS buffer.
// ---------------------------------------------------------------------------
__global__ void enc_kernel(const float* __restrict__ x, const float* __restrict__ u,
                           const int* __restrict__ batch,
                           const float* __restrict__ W1, const float* __restrict__ b1,
                           const float* __restrict__ a1,
                           const float* __restrict__ W2, const float* __restrict__ b2,
                           const float* __restrict__ a2,
                           const float* __restrict__ Wf, const float* __restrict__ bf,
                           float* __restrict__ hout) {
  __shared__ float act[4][16 * HID];
  const int lane = threadIdx.x & 31;
  const int wv   = threadIdx.x >> 5;
  int tile = blockIdx.x * 4 + wv;
  if (tile >= NTILE_N) tile = NTILE_N - 1;   // duplicate tail work (benign)
  const int node0 = tile * 16;
  const int row   = node0 + (lane & 15);
  const int bidx  = batch[row];
  const float s1 = a1[0], s2 = a2[0];
  float* my = act[wv];

  // ---- layer 1: 24 -> 128 ----
  v8f acc[8];
#pragma unroll
  for (int nt = 0; nt < 8; ++nt) acc[nt] = bcast8(b1[nt * 16 + (lane & 15)]);
#pragma unroll
  for (int ks = 0; ks < 6; ++ks) {
    const int k0 = ks * 4;
    const int kk = k0 + 2 * (lane >> 4);
    v2f a;
    if (kk < 16) { float2 t = *(const float2*)(x + (size_t)row * 16 + kk);        a.x = t.x; a.y = t.y; }
    else         { float2 t = *(const float2*)(u + (size_t)bidx * 8 + (kk - 16)); a.x = t.x; a.y = t.y; }
#pragma unroll
    for (int nt = 0; nt < 8; ++nt)
      acc[nt] = wmma4(a, bfrag(W1, HID, k0, nt * 16, lane), acc[nt]);
  }
#pragma unroll
  for (int nt = 0; nt < 8; ++nt)
#pragma unroll
    for (int r = 0; r < 8; ++r)
      my[(r + 8 * (lane >> 4)) * HID + nt * 16 + (lane & 15)] = prelu(acc[nt][r], s1);
  __syncthreads();

  // ---- layer 2: 128 -> 128 ----
#pragma unroll
  for (int nt = 0; nt < 8; ++nt) acc[nt] = bcast8(b2[nt * 16 + (lane & 15)]);
  for (int ks = 0; ks < 32; ++ks) {
    const int k0 = ks * 4;
    const int kk = k0 + 2 * (lane >> 4);
    v2f a; a.x = my[(lane & 15) * HID + kk]; a.y = my[(lane & 15) * HID + kk + 1];
#pragma unroll
    for (int nt = 0; nt < 8; ++nt)
      acc[nt] = wmma4(a, bfrag(W2, HID, k0, nt * 16, lane), acc[nt]);
  }
  __syncthreads();
#pragma unroll
  for (int nt = 0; nt < 8; ++nt)
#pragma unroll
    for (int r = 0; r < 8; ++r)
      my[(r + 8 * (lane >> 4)) * HID + nt * 16 + (lane & 15)] = prelu(acc[nt][r], s2);
  __syncthreads();

  // ---- final: 128 -> 64 ----
  v8f fac[4];
#pragma unroll
  for (int nt = 0; nt < 4; ++nt) fac[nt] = bcast8(bf[nt * 16 + (lane & 15)]);
  for (int ks = 0; ks < 32; ++ks) {
    const int k0 = ks * 4;
    const int kk = k0 + 2 * (lane >> 4);
    v2f a; a.x = my[(lane & 15) * HID + kk]; a.y = my[(lane & 15) * HID + kk + 1];
#pragma unroll
    for (int nt = 0; nt < 4; ++nt)
      fac[nt] = wmma4(a, bfrag(Wf, EMB, k0, nt * 16, lane), fac[nt]);
  }
#pragma unroll
  for (int nt = 0; nt < 4; ++nt)
#pragma unroll
    for (int r = 0; r < 8; ++r)
      hout[(size_t)(node0 + r + 8 * (lane >> 4)) * EMB + nt * 16 + (lane & 15)] = fac[nt][r];
}

// ---------------------------------------------------------------------------
// Zero agg: 3125 blocks * 256 threads * float4 == NN*64 floats exactly.
// ---------------------------------------------------------------------------
__global__ void zero_kernel(float4* __restrict__ p) {
  p[blockIdx.x * 256 + threadIdx.x] = make_float4(0.f, 0.f, 0.f, 0.f);
}

// ---------------------------------------------------------------------------
// Scatter: agg[dst[e], :] += h[src[e], :]. One thread = one (edge, float2).
// ---------------------------------------------------------------------------
__global__ void scatter_kernel(const float* __restrict__ h,
                               const int* __restrict__ src, const int* __restrict__ dst,
                               float* __restrict__ agg) {
  const int i = blockIdx.x * 256 + threadIdx.x;     // [0, EE*32)
  const int e = i >> 5;
  const int c = (i & 31) * 2;
  const int s = src[e], d = dst[e];
  float2 v = *(const float2*)(h + (size_t)s * EMB + c);
  atomicAdd(agg + (size_t)d * EMB + c,     v.x);
  atomicAdd(agg + (size_t)d * EMB + c + 1, v.y);
}

// ---------------------------------------------------------------------------
// SAGEConv update: hnext = agg @ Wl + bl + hcur @ Wr   (both K=64 GEMMs fused)
// 256 threads = 8 waves; wave = one 16-node tile; no LDS needed.
// ---------------------------------------------------------------------------
__global__ void conv_kernel(const float* __restrict__ agg, const float* __restrict__ hcur,
                            const float* __restrict__ Wl, const float* __restrict__ bl,
                            const float* __restrict__ Wr, float* __restrict__ hnext) {
  const int lane = threadIdx.x & 31;
  const int wv   = threadIdx.x >> 5;
  int tile = blockIdx.x * 8 + wv;
  if (tile >= NTILE_N) tile = NTILE_N - 1;
  const int node0 = tile * 16;

  v8f acc[4];
#pragma unroll
  for (int nt = 0; nt < 4; ++nt) acc[nt] = bcast8(bl[nt * 16 + (lane & 15)]);
  for (int ks = 0; ks < 16; ++ks) {
    const int k0 = ks * 4;
    v2f aa = afrag_g(agg,  EMB, node0, k0, lane);
    v2f ah = afrag_g(hcur, EMB, node0, k0, lane);
#pragma unroll
    for (int nt = 0; nt < 4; ++nt) {
      acc[nt] = wmma4(aa, bfrag(Wl, EMB, k0, nt * 16, lane), acc[nt]);
      acc[nt] = wmma4(ah, bfrag(Wr, EMB, k0, nt * 16, lane), acc[nt]);
    }
  }
#pragma unroll
  for (int nt = 0; nt < 4; ++nt)
#pragma unroll
    for (int r = 0; r < 8; ++r)
      hnext[(size_t)(node0 + r + 8 * (lane >> 4)) * EMB + nt * 16 + (lane & 15)] = acc[nt][r];
}

// ---------------------------------------------------------------------------
// Decoder: out[m,3] = MLP(h[real_nodes[m]])  (64->128->128->3)
// 128 threads = 4 waves; wave = one 16-row tile (tail tile guarded).
// ---------------------------------------------------------------------------
__global__ void dec_kernel(const float* __restrict__ h, const int* __restrict__ rn,
                           const float* __restrict__ W1, const float* __restrict__ b1,
                           const float* __restrict__ a1,
                           const float* __restrict__ W2, const float* __restrict__ b2,
                           const float* __restrict__ a2,
                           const float* __restrict__ Wf, const float* __restrict__ bf,
                           float* __restrict__ out) {
  __shared__ float act[4][16 * HID];
  const int lane = threadIdx.x & 31;
  const int wv   = threadIdx.x >> 5;
  int tile = blockIdx.x * 4 + wv;
  if (tile >= NTILE_M) tile = NTILE_M - 1;
  const int grow = tile * 16 + (lane & 15);
  const int gidx = rn[grow < MREAL ? grow : MREAL - 1];
  const float s1 = a1[0], s2 = a2[0];
  float* my = act[wv];

  // ---- layer 1: 64 -> 128 ----
  v8f acc[8];
#pragma unroll
  for (int nt = 0; nt < 8; ++nt) acc[nt] = bcast8(b1[nt * 16 + (lane & 15)]);
  for (int ks = 0; ks < 16; ++ks) {
    const int k0 = ks * 4;
    const int kk = k0 + 2 * (lane >> 4);
    float2 t = *(const float2*)(h + (size_t)gidx * EMB + kk);
    v2f a; a.x = t.x; a.y = t.y;
#pragma unroll
    for (int nt = 0; nt < 8; ++nt)
      acc[nt] = wmma4(a, bfrag(W1, HID, k0, nt * 16, lane), acc[nt]);
  }
#pragma unroll
  for (int nt = 0; nt < 8; ++nt)
#pragma unroll
    for (int r = 0; r < 8; ++r)
      my[(r + 8 * (lane >> 4)) * HID + nt * 16 + (lane & 15)] = prelu(acc[nt][r], s1);
  __syncthreads();

  // ---- layer 2: 128 -> 128 ----
#pragma unroll
  for (int nt = 0; nt < 8; ++nt) acc[nt] = bcast8(b2[nt * 16 + (lane & 15)]);
  for (int ks = 0; ks < 32; ++ks) {
    const int k0 = ks * 4;
    const int kk = k0 + 2 * (lane >> 4);
    v2f a; a.x = my[(lane & 15) * HID + kk]; a.y = my[(lane & 15) * HID + kk + 1];
#pragma unroll
    for (int nt = 0; nt < 8; ++nt)
      acc[nt] = wmma4(a, bfrag(W2, HID, k0, nt * 16, lane), acc[nt]);
  }
  __syncthreads();
#pragma unroll
  for (int nt = 0; nt < 8; ++nt)
#pragma unroll
    for (int r = 0; r < 8; ++r)
      my[(r + 8 * (lane >> 4)) * HID + nt * 16 + (lane & 15)] = prelu(acc[nt][r], s2);
  __syncthreads();

  // ---- final: 128 -> 3 (48 dot products per wave, scalar) ----
  for (int idx = lane; idx < 48; idx += 32) {
    const int row = idx / 3, c = idx % 3;
    float sum = bf[c];
#pragma unroll 8
    for (int k = 0; k < HID; ++k) sum += my[row * HID + k] * Wf[k * 3 + c];
    const int orow = tile * 16 + row;
    if (orow < MREAL) out[(size_t)orow * 3 + c] = sum;
  }
}

// ---------------------------------------------------------------------------
// Host launcher. Input order = setup_inputs() dict order with params expanded
// in jax tree_leaves order (sorted keys: convs < dec < enc; per-layer W<a<b):
//  0:x 1:u
//  2..10:  conv{0,1,2}.{Wl,Wr,bl}
//  11..18: dec.final.{W,b}, dec.h0.{W,a,b}, dec.h1.{W,a,b}
//  19..26: enc.final.{W,b}, enc.h0.{W,a,b}, enc.h1.{W,a,b}
//  27:batch 28:edge_index[2,E] 29:real_nodes
// ---------------------------------------------------------------------------
extern "C" void kernel_launch(void* const* d_in, const int* in_sizes, int n_in,
                              void* d_out, int out_size, void* d_ws, size_t ws_size,
                              hipStream_t stream) {
  const float* x = (const float*)d_in[0];
  const float* u = (const float*)d_in[1];
  const float* cWl[3], *cWr[3], *cbl[3];
  for (int r = 0; r < 3; ++r) {
    cWl[r] = (const float*)d_in[2 + 3 * r];
    cWr[r] = (const float*)d_in[3 + 3 * r];
    cbl[r] = (const float*)d_in[4 + 3 * r];
  }
  const float* dWf = (const float*)d_in[11];
  const float* dbf = (const float*)d_in[12];
  const float* dW1 = (const float*)d_in[13];
  const float* da1 = (const float*)d_in[14];
  const float* db1 = (const float*)d_in[15];
  const float* dW2 = (const float*)d_in[16];
  const float* da2 = (const float*)d_in[17];
  const float* db2 = (const float*)d_in[18];
  const float* eWf = (const float*)d_in[19];
  const float* ebf = (const float*)d_in[20];
  const float* eW1 = (const float*)d_in[21];
  const float* ea1 = (const float*)d_in[22];
  const float* eb1 = (const float*)d_in[23];
  const float* eW2 = (const float*)d_in[24];
  const float* ea2 = (const float*)d_in[25];
  const float* eb2 = (const float*)d_in[26];
  const int* batch = (const int*)d_in[27];
  const int* src   = (const int*)d_in[28];        // edge_index[0]
  const int* dst   = (const int*)d_in[28] + EE;   // edge_index[1]
  const int* rn    = (const int*)d_in[29];
  float* out = (float*)d_out;

  float* h0  = (float*)d_ws;
  float* h1  = h0 + (size_t)NN * EMB;
  float* agg = h1 + (size_t)NN * EMB;

  // encoder: 3125 tiles, 4 waves/block
  enc_kernel<<<(NTILE_N + 3) / 4, 128, 0, stream>>>(
      x, u, batch, eW1, eb1, ea1, eW2, eb2, ea2, eWf, ebf, h0);

  float* cur = h0; float* nxt = h1;
  for (int r = 0; r < 3; ++r) {
    zero_kernel<<<3125, 256, 0, stream>>>((float4*)agg);              // NN*64 floats
    scatter_kernel<<<(EE * 32) / 256, 256, 0, stream>>>(cur, src, dst, agg);
    conv_kernel<<<(NTILE_N + 7) / 8, 256, 0, stream>>>(agg, cur, cWl[r], cbl[r], cWr[r], nxt);
    float* t = cur; cur = nxt; nxt = t;
  }

  // decoder: 1563 tiles, 4 waves/block
  dec_kernel<<<(NTILE_M + 3) / 4, 128, 0, stream>>>(
      cur, rn, dW1, db1, da1, dW2, db2, da2, dWf, dbf, out);
}